// GCN_54752243089878
// MI455X (gfx1250) — compile-verified
//
#include <hip/hip_runtime.h>

#define NN 100000
#define NE 1600000

typedef float v2f __attribute__((ext_vector_type(2)));
typedef float v8f __attribute__((ext_vector_type(8)));

// ---------------- degree / norm precompute ----------------
__global__ __launch_bounds__(256) void k_deg(const int* __restrict__ dst,
                                             unsigned* __restrict__ deg, int E) {
  int e = blockIdx.x * 256 + threadIdx.x;
  if (e < E) atomicAdd(&deg[dst[e]], 1u);
}

__global__ __launch_bounds__(256) void k_dinv(const unsigned* __restrict__ deg,
                                              float* __restrict__ dinv, int N) {
  int i = blockIdx.x * 256 + threadIdx.x;
  if (i < N) dinv[i] = 1.0f / sqrtf((float)deg[i] + 1.0f);  // +1 self-loop
}

__global__ __launch_bounds__(256) void k_norm(const int* __restrict__ src,
                                              const int* __restrict__ dst,
                                              const float* __restrict__ dinv,
                                              float* __restrict__ nrm, int E) {
  int e = blockIdx.x * 256 + threadIdx.x;
  if (e < E) nrm[e] = dinv[src[e]] * dinv[dst[e]];
}

// ---------------- GEMM: H[M x 64] = X[M x K] @ W[K x 64] via fp32 WMMA ----
// One wave per 16-row strip; 4 accumulators cover N=64.
// W staged in LDS pair-interleaved: wl[p*64+n] = {W[2p][n], W[2p+1][n]}
// so each B fragment is a single aligned ds_load_b64 (no VGPR repacking).
template <int K>
__global__ __launch_bounds__(256) void k_gemm(const float* __restrict__ X,
                                              const float* __restrict__ W,
                                              float* __restrict__ H, int M) {
  __shared__ __align__(16) v2f wl[(K / 2) * 64];
  for (int i = threadIdx.x; i < (K / 2) * 64; i += 256) {
    int p = i >> 6, n = i & 63;
    v2f t = { W[(size_t)(2 * p) * 64 + n], W[(size_t)(2 * p + 1) * 64 + n] };
    wl[i] = t;
  }
  __syncthreads();

  int wave  = threadIdx.x >> 5;
  int lane  = threadIdx.x & 31;
  int strip = blockIdx.x * 8 + wave;
  if (strip * 16 >= M) return;
  int row0 = strip * 16;
  int m    = lane & 15;   // row within tile (A) / column (B,C)
  int h2   = lane >> 4;   // half-wave selects K pair {0,1} vs {2,3}

  const float* xrow = X + (size_t)(row0 + m) * K;
  v8f a0 = {}, a1 = {}, a2 = {}, a3 = {};
  for (int kb = 0; kb < K; kb += 4) {
    int pa = (kb >> 1) + h2;                  // k-pair index for this half-wave
    v2f a = *(const v2f*)(xrow + 2 * pa);     // A[m][2pa], A[m][2pa+1]
    const v2f* wp = &wl[pa * 64 + m];
    v2f b0 = wp[0];
    v2f b1 = wp[16];
    v2f b2 = wp[32];
    v2f b3 = wp[48];
    a0 = __builtin_amdgcn_wmma_f32_16x16x4_f32(false, a, false, b0, (short)0, a0, false, false);
    a1 = __builtin_amdgcn_wmma_f32_16x16x4_f32(false, a, false, b1, (short)0, a1, false, false);
    a2 = __builtin_amdgcn_wmma_f32_16x16x4_f32(false, a, false, b2, (short)0, a2, false, false);
    a3 = __builtin_amdgcn_wmma_f32_16x16x4_f32(false, a, false, b3, (short)0, a3, false, false);
  }
  // C/D layout: VGPR j holds row (j + 8*h2), column m
  float* out0 = H + (size_t)(row0 + 8 * h2) * 64 + m;
#pragma unroll
  for (int j = 0; j < 8; ++j) {
    float* r = out0 + (size_t)j * 64;
    r[0] = a0[j]; r[16] = a1[j]; r[32] = a2[j]; r[48] = a3[j];
  }
}

// ---------------- edge aggregation: agg[dst] += norm * h[src] -------------
__global__ __launch_bounds__(256) void k_agg(const float* __restrict__ H,
                                             const int* __restrict__ src,
                                             const int* __restrict__ dst,
                                             const float* __restrict__ nrm,
                                             float* __restrict__ agg, int E) {
  int e = (int)((blockIdx.x * 256 + threadIdx.x) >> 5);
  if (e >= E) return;
  int lane = threadIdx.x & 31;
  int s = src[e], d = dst[e];
  float w = nrm[e];
  v2f hv = *(const v2f*)(H + (size_t)s * 64 + lane * 2);
  float* p = agg + (size_t)d * 64 + lane * 2;
  unsafeAtomicAdd(p,     w * hv.x);
  unsafeAtomicAdd(p + 1, w * hv.y);
}

// ---------------- epilogue: +self-loop +bias, relu, L2 row-normalize ------
__global__ __launch_bounds__(256) void k_relu_l2(const float* __restrict__ agg,
                                                 const float* __restrict__ H,
                                                 const float* __restrict__ dinv,
                                                 const float* __restrict__ bias,
                                                 float* __restrict__ out, int N) {
  int node = (int)((blockIdx.x * 256 + threadIdx.x) >> 5);
  if (node >= N) return;
  int lane = threadIdx.x & 31;
  float di = dinv[node];
  float sl = di * di;
  size_t base = (size_t)node * 64 + lane * 2;
  v2f a  = *(const v2f*)(agg + base);
  v2f hv = *(const v2f*)(H + base);
  float vx = fmaxf(a.x + sl * hv.x + bias[lane * 2],     0.0f);
  float vy = fmaxf(a.y + sl * hv.y + bias[lane * 2 + 1], 0.0f);
  float ss = vx * vx + vy * vy;
#pragma unroll
  for (int off = 16; off > 0; off >>= 1) ss += __shfl_xor(ss, off, 32);
  float scale = 1.0f / fmaxf(sqrtf(ss), 1e-12f);
  out[base]     = vx * scale;
  out[base + 1] = vy * scale;
}

// ---------------- final epilogue: +self-loop +bias (no relu/norm) ---------
__global__ __launch_bounds__(256) void k_final(const float* __restrict__ agg,
                                               const float* __restrict__ H,
                                               const float* __restrict__ dinv,
                                               const float* __restrict__ bias,
                                               float* __restrict__ out, int N) {
  int idx = blockIdx.x * 256 + threadIdx.x;
  if (idx >= N * 64) return;
  int node = idx >> 6;
  int c    = idx & 63;
  float di = dinv[node];
  out[idx] = agg[idx] + di * di * H[idx] + bias[c];
}

extern "C" void kernel_launch(void* const* d_in, const int* in_sizes, int n_in,
                              void* d_out, int out_size, void* d_ws, size_t ws_size,
                              hipStream_t stream) {
  const float* x  = (const float*)d_in[0];
  const int*   ei = (const int*)d_in[1];          // [2, NE] int32
  const float* W1 = (const float*)d_in[2];
  const float* b1 = (const float*)d_in[3];
  const float* W2 = (const float*)d_in[4];
  const float* b2 = (const float*)d_in[5];
  const float* W3 = (const float*)d_in[6];
  const float* b3 = (const float*)d_in[7];
  float* out = (float*)d_out;

  const int* src = ei;
  const int* dst = ei + NE;

  // workspace bump allocator (256B aligned)
  char* ws = (char*)d_ws;
  size_t off = 0;
  auto carve = [&](size_t bytes) {
    void* p = ws + off;
    off = (off + bytes + 255) & ~(size_t)255;
    return p;
  };
  unsigned* deg  = (unsigned*)carve((size_t)NN * 4);
  float*    dinv = (float*)   carve((size_t)NN * 4);
  float*    nrm  = (float*)   carve((size_t)NE * 4);
  float*    h    = (float*)   carve((size_t)NN * 64 * 4);
  float*    agg  = (float*)   carve((size_t)NN * 64 * 4);
  float*    hn   = (float*)   carve((size_t)NN * 64 * 4);

  const int eBlk   = (NE + 255) / 256;          // per-edge kernels
  const int nBlk   = (NN + 255) / 256;          // per-node kernels
  const int gBlk   = ((NN / 16) + 7) / 8;       // gemm: 8 strips/block
  const int aggBlk = NE / 8;                    // agg: 8 edges(waves)/block
  const int wnBlk  = (NN * 32 + 255) / 256;     // wave-per-node kernels
  const int feBlk  = (NN * 64 + 255) / 256;     // elementwise final

  // ---- precompute deg / dinv / norm ----
  hipMemsetAsync(deg, 0, (size_t)NN * 4, stream);
  k_deg <<<eBlk, 256, 0, stream>>>(dst, deg, NE);
  k_dinv<<<nBlk, 256, 0, stream>>>(deg, dinv, NN);
  k_norm<<<eBlk, 256, 0, stream>>>(src, dst, dinv, nrm, NE);

  // ---- layer 1 ----
  k_gemm<256><<<gBlk, 256, 0, stream>>>(x, W1, h, NN);
  hipMemsetAsync(agg, 0, (size_t)NN * 64 * 4, stream);
  k_agg    <<<aggBlk, 256, 0, stream>>>(h, src, dst, nrm, agg, NE);
  k_relu_l2<<<wnBlk,  256, 0, stream>>>(agg, h, dinv, b1, hn, NN);

  // ---- layer 2 ----
  k_gemm<64><<<gBlk, 256, 0, stream>>>(hn, W2, h, NN);
  hipMemsetAsync(agg, 0, (size_t)NN * 64 * 4, stream);
  k_agg    <<<aggBlk, 256, 0, stream>>>(h, src, dst, nrm, agg, NE);
  k_relu_l2<<<wnBlk,  256, 0, stream>>>(agg, h, dinv, b2, hn, NN);

  // ---- layer 3 ----
  k_gemm<64><<<gBlk, 256, 0, stream>>>(hn, W3, h, NN);
  hipMemsetAsync(agg, 0, (size_t)NN * 64 * 4, stream);
  k_agg  <<<aggBlk, 256, 0, stream>>>(h, src, dst, nrm, agg, NE);
  k_final<<<feBlk,  256, 0, stream>>>(agg, h, dinv, b3, out, NN);

  (void)in_sizes; (void)n_in; (void)out_size; (void)ws_size;
}